// SemiCRFShiftRelay_80255758893450
// MI455X (gfx1250) — compile-verified
//
#include <hip/hip_runtime.h>
#include <stdint.h>

// Problem constants (match reference)
#define BB 256
#define TT 2048
#define LL 64

#define WAVES_PER_BLOCK 8
#define CHUNK_T 2                 // timesteps per async chunk (2*64*4 = 512 B = 32 lanes x 16 B)
#define RING_CHUNKS 8             // LDS ring depth per wave (4 KB)
#define PREFETCH 6                // async chunks in flight
#define NCHUNKS (TT / CHUNK_T)    // 1024

__device__ __forceinline__ float wave_max(float v) {
#pragma unroll
  for (int off = 16; off > 0; off >>= 1)
    v = fmaxf(v, __shfl_xor(v, off, 32));
  return v;
}

__device__ __forceinline__ float wave_sum(float v) {
#pragma unroll
  for (int off = 16; off > 0; off >>= 1)
    v += __shfl_xor(v, off, 32);
  return v;
}

__global__ __launch_bounds__(WAVES_PER_BLOCK * 32) void semicrf_scan(
    const float* __restrict__ logits,        // (B,T,L) f32
    const float* __restrict__ relay_logits,  // (B,T)   f32
    const int* __restrict__ relay_target,    // (B,T)   i32
    const uint8_t* __restrict__ relay_mask,  // (B,T)   bool (1B)
    const uint8_t* __restrict__ end_mask,    // (B,T)   bool (1B)
    const int* __restrict__ seq_len,         // (B,)    i32
    float* __restrict__ out)                 // (B,)    f32
{
  __shared__ float ring[WAVES_PER_BLOCK][RING_CHUNKS * CHUNK_T * LL];

  const int lane = (int)(threadIdx.x & 31u);
  const int wid = (int)(threadIdx.x >> 5);
  const int batch = (int)blockIdx.x * WAVES_PER_BLOCK + wid;

  const float NEG_INF = -__builtin_inff();
  const float* gbase = logits + (size_t)batch * TT * LL;

  // CDNA5 flat->LDS: generic pointer low 32 bits are the LDS byte address.
  const uint32_t ring_base = (uint32_t)(uintptr_t)(&ring[wid][0]);

  // Issue one async chunk (2 timesteps of logits) into the per-wave LDS ring.
  auto issue_chunk = [&](int c) {
    uint64_t g = (uint64_t)(uintptr_t)(gbase + (size_t)c * (CHUNK_T * LL)) +
                 (uint64_t)(lane * 16);
    uint32_t l = ring_base +
                 (uint32_t)((c & (RING_CHUNKS - 1)) * (CHUNK_T * LL * 4) + lane * 16);
    asm volatile("global_load_async_to_lds_b128 %0, %1, off"
                 :: "v"(l), "v"(g)
                 : "memory");
  };

#pragma unroll
  for (int c = 0; c < PREFETCH; ++c) issue_chunk(c);

  // ---- Initialization (t = 1 state) ----
  const size_t bt = (size_t)batch * TT;
  const float l00 = gbase[0];            // logits[b,0,0]
  const float rl0 = relay_logits[bt];    // relay_logits[b,0]

  // Register-resident histories, striped one element per lane:
  //   s_a[lane j] = scores[t - j]        s_b[lane j] = scores[t - 32 - j]
  //   r_a[lane j] = relay[t - 1 - j]     r_b[lane j] = relay[t - 33 - j]
  float s_a = (lane == 0) ? l00 : ((lane == 1) ? 0.0f : NEG_INF);
  float s_b = NEG_INF;
  float r_a = (lane == 0) ? rl0 : NEG_INF;
  float r_b = NEG_INF;

  // gold path: t = 0 term (relay_target forced to 0 at t=0)
  float gold = rl0 * (float)relay_mask[bt] + l00 * (float)end_mask[bt];

  const int myseq = seq_len[batch];  // in [65, 2048]
  float saved = 0.0f;

  // metadata staged 32 timesteps at a time (T % 32 == 0, so no OOB)
  float rl_blk = 0.0f;
  int rt_blk = 0, mk_blk = 0, em_blk = 0;

  for (int c = 0; c < NCHUNKS; ++c) {
    const int t0 = c * CHUNK_T;
    if ((t0 & 31) == 0) {
      const size_t mi = bt + (size_t)(t0 + lane);
      rl_blk = relay_logits[mi];
      rt_blk = relay_target[mi];
      mk_blk = (int)relay_mask[mi];
      em_blk = (int)end_mask[mi];
    }

    // oldest in-flight chunk (c) is complete once <= PREFETCH-1 remain
    asm volatile("s_wait_asynccnt 5" ::: "memory");

    const float* lchunk = &ring[wid][(c & (RING_CHUNKS - 1)) * (CHUNK_T * LL)];
    const float lt_lo0 = lchunk[lane];
    const float lt_hi0 = lchunk[lane + 32];
    const float lt_lo1 = lchunk[LL + lane];
    const float lt_hi1 = lchunk[LL + lane + 32];

    if (c + PREFETCH < NCHUNKS) issue_chunk(c + PREFETCH);

#pragma unroll
    for (int k = 0; k < CHUNK_T; ++k) {
      const int t = t0 + k;
      if (t == 0) continue;  // scan starts at t = 1
      const float lo = k ? lt_lo1 : lt_lo0;  // logits[b,t,lane]
      const float hi = k ? lt_hi1 : lt_hi0;  // logits[b,t,lane+32]

      // segment candidates: length l = lane+1 and l = lane+33
      const float v1 = s_a + lo;
      const float v2 = s_b + hi;
      // relay-terminated length-64 segment: relay[t-64] lives in r_b @ lane 31
      const float v3 = (lane == 31) ? (r_b + hi) : NEG_INF;

      const float m = wave_max(fmaxf(v1, fmaxf(v2, v3)));
      const float s =
          wave_sum(__expf(v1 - m) + __expf(v2 - m) + __expf(v3 - m));
      const float new_score = m + __logf(s);  // scores[t+1], uniform over wave

      // gold path accumulation (all operands lane-uniform after broadcasts)
      const int sel = t & 31;
      const float rl = __shfl(rl_blk, sel, 32);
      const int rt = __shfl(rt_blk, sel, 32);
      const float mk = (float)__shfl(mk_blk, sel, 32);
      const float em = (float)__shfl(em_blk, sel, 32);
      const float seg = (rt < 32) ? __shfl(lo, rt & 31, 32)
                                  : __shfl(hi, rt & 31, 32);
      gold += rl * mk + seg * em;

      // relay recurrence: relay[t] = rl + logaddexp(scores[t], relay[t-1])
      // (lane 0 holds both operands; other lanes produce garbage, never used)
      const float st = s_a;   // lane 0: scores[t]
      const float rp = r_a;   // lane 0: relay[t-1]
      const float mm = fmaxf(st, rp);
      const float new_relay =
          (t < TT - LL) ? (rl + mm + __logf(__expf(st - mm) + __expf(rp - mm)))
                        : 0.0f;

      if (t + 1 == myseq) saved = new_score;

      // shift histories one lane; insert new values at lane 0
      const float a31 = __shfl(s_a, 31, 32);  // scores[t-31]
      const float r31 = __shfl(r_a, 31, 32);  // relay[t-32]
      s_a = __shfl_up(s_a, 1, 32); if (lane == 0) s_a = new_score;
      s_b = __shfl_up(s_b, 1, 32); if (lane == 0) s_b = a31;
      r_a = __shfl_up(r_a, 1, 32); if (lane == 0) r_a = new_relay;
      r_b = __shfl_up(r_b, 1, 32); if (lane == 0) r_b = r31;
    }
  }

  if (lane == 0) out[batch] = saved - gold;
}

extern "C" void kernel_launch(void* const* d_in, const int* in_sizes, int n_in,
                              void* d_out, int out_size, void* d_ws,
                              size_t ws_size, hipStream_t stream) {
  (void)in_sizes; (void)n_in; (void)out_size; (void)d_ws; (void)ws_size;
  const float* logits = (const float*)d_in[0];
  const float* relay_logits = (const float*)d_in[1];
  const int* relay_target = (const int*)d_in[2];
  const uint8_t* relay_mask = (const uint8_t*)d_in[3];
  const uint8_t* end_mask = (const uint8_t*)d_in[4];
  const int* seq_len = (const int*)d_in[5];
  float* out = (float*)d_out;

  dim3 grid(BB / WAVES_PER_BLOCK);
  dim3 block(WAVES_PER_BLOCK * 32);
  semicrf_scan<<<grid, block, 0, stream>>>(logits, relay_logits, relay_target,
                                           relay_mask, end_mask, seq_len, out);
}